// TopKSparsemaxWrapper_24309514895544
// MI455X (gfx1250) — compile-verified
//
#include <hip/hip_runtime.h>
#include <math.h>

#define BATCH  4096
#define LATENT 4096
#define KSEL   10

typedef float v2f __attribute__((ext_vector_type(2)));
typedef float v4f __attribute__((ext_vector_type(4)));
typedef float v8f __attribute__((ext_vector_type(8)));

// Branch-free insert of packed (cost<<32 | idx) key into ascending sorted 10-array.
// Lexicographic (cost, idx) ordering is exact since cost >= 0 (monotonic as uint bits).
__device__ __forceinline__ void ins10(unsigned long long a[10], unsigned long long v) {
  unsigned long long cur = v;
#pragma unroll
  for (int i = 0; i < 10; ++i) {
    const bool gt = a[i] > cur;
    const unsigned long long hi = gt ? a[i] : cur;
    const unsigned long long lo = gt ? cur : a[i];
    a[i] = lo;
    cur  = hi;
  }
}

__device__ __forceinline__ unsigned long long wave_min_ull(unsigned long long v) {
#pragma unroll
  for (int off = 16; off > 0; off >>= 1) {
    const unsigned long long o = __shfl_xor(v, off, 32);
    v = (o < v) ? o : v;
  }
  return v;
}

// Merge per-lane sorted top-10 arrays into the wave-global 10 smallest (ascending).
// All keys are unique (index in low bits), so equality identifies the winning lane.
__device__ __forceinline__ void merge10(unsigned long long a[10], unsigned long long out[10]) {
#pragma unroll
  for (int c = 0; c < 10; ++c) {
    const unsigned long long best = wave_min_ull(a[0]);
    out[c] = best;
    const bool adv = (a[0] == best);
#pragma unroll
    for (int i = 0; i < 9; ++i) a[i] = adv ? a[i + 1] : a[i];
    a[9] = adv ? ~0ULL : a[9];
  }
}

// One wave per row: top-10 smallest |logit|, 1024 subset sums via WMMA outer-sum,
// 10 best masks, sparsemax, entropy, metadata for the bv writer.
__global__ void __launch_bounds__(256) topk_rows_kernel(const float* __restrict__ logits,
                                                        float* __restrict__ distr,
                                                        float* __restrict__ ent,
                                                        int* __restrict__ meta) {
  const int lane = threadIdx.x & 31;
  const int row  = (int)((blockIdx.x * blockDim.x + threadIdx.x) >> 5);
  if (row >= BATCH) return;
  const float* __restrict__ r = logits + (size_t)row * LATENT;

  unsigned long long a[10];
#pragma unroll
  for (int i = 0; i < 10; ++i) a[i] = ~0ULL;
  float spos = 0.f;

#pragma unroll 4
  for (int i = 0; i < LATENT / (32 * 4); ++i) {
    const int j = (i * 32 + lane) * 4;
    const v4f x = *reinterpret_cast<const v4f*>(r + j);
    {
      const float v = x.x; spos += (v > 0.f) ? v : 0.f;
      ins10(a, ((unsigned long long)__float_as_uint(fabsf(v)) << 32) | (unsigned int)(j + 0));
    }
    {
      const float v = x.y; spos += (v > 0.f) ? v : 0.f;
      ins10(a, ((unsigned long long)__float_as_uint(fabsf(v)) << 32) | (unsigned int)(j + 1));
    }
    {
      const float v = x.z; spos += (v > 0.f) ? v : 0.f;
      ins10(a, ((unsigned long long)__float_as_uint(fabsf(v)) << 32) | (unsigned int)(j + 2));
    }
    {
      const float v = x.w; spos += (v > 0.f) ? v : 0.f;
      ins10(a, ((unsigned long long)__float_as_uint(fabsf(v)) << 32) | (unsigned int)(j + 3));
    }
  }
#pragma unroll
  for (int off = 16; off > 0; off >>= 1) spos += __shfl_xor(spos, off, 32);

  unsigned long long top[10];
  merge10(a, top);

  float csm[10];
  int   pos[10];
#pragma unroll
  for (int c = 0; c < 10; ++c) {
    csm[c] = __uint_as_float((unsigned int)(top[c] >> 32));
    pos[c] = (int)(top[c] & 0xffffffffu);
  }

  // Subset sums: sum(m) = s_lo[m&15] + s_hi[m>>4]. Outer-sum as rank-2 matmul:
  // D(16x16) = A(16x4) x B(4x16), A = [s_lo | 1 | 0 | 0], B = [1 ; s_hi ; 0 ; 0].
  float slo = 0.f;
  {
    const int rr = lane & 15;
#pragma unroll
    for (int k = 0; k < 4; ++k) if ((rr >> k) & 1) slo += csm[k];
  }
  v2f A;
  A.x = (lane < 16) ? slo : 0.f;   // VGPR0: lanes0-15 K=0 (s_lo), lanes16-31 K=2 (0)
  A.y = (lane < 16) ? 1.f : 0.f;   // VGPR1: lanes0-15 K=1 (ones), lanes16-31 K=3 (0)

  unsigned long long sa[10];
#pragma unroll
  for (int i = 0; i < 10; ++i) sa[i] = ~0ULL;

#pragma unroll
  for (int t = 0; t < 4; ++t) {
    const int q = t * 16 + (lane & 15);
    float shi = 0.f;
#pragma unroll
    for (int k = 0; k < 6; ++k) if ((q >> k) & 1) shi += csm[4 + k];
    v2f Bv;
    Bv.x = (lane < 16) ? 1.f : 0.f;  // K=0 row = ones
    Bv.y = (lane < 16) ? shi : 0.f;  // K=1 row = s_hi tile
    v8f C = {0.f, 0.f, 0.f, 0.f, 0.f, 0.f, 0.f, 0.f};
    v8f D = __builtin_amdgcn_wmma_f32_16x16x4_f32(false, A, false, Bv, (short)0, C, false, false);
    const int Ng = t * 16 + (lane & 15);
    const int Mb = (lane >= 16) ? 8 : 0;
#pragma unroll
    for (int v = 0; v < 8; ++v) {
      const int m = (v + Mb) + 16 * Ng;  // low 4 bits = M, high 6 bits = N
      ins10(sa, ((unsigned long long)__float_as_uint(D[v]) << 32) | (unsigned int)m);
    }
  }

  unsigned long long best[10];
  merge10(sa, best);

  if (lane == 0) {
    float z[10], cum[10];
    float csum = 0.f;
#pragma unroll
    for (int c = 0; c < 10; ++c) {
      z[c] = spos - __uint_as_float((unsigned int)(best[c] >> 32));  // descending
      csum += z[c];
      cum[c] = csum;
    }
    int kk = 0;
#pragma unroll
    for (int c = 0; c < 10; ++c)
      if (1.f + (float)(c + 1) * z[c] > cum[c]) kk++;
    float csel = cum[0];
#pragma unroll
    for (int c = 1; c < 10; ++c) csel = (kk - 1 == c) ? cum[c] : csel;
    const float tau = (csel - 1.f) / (float)kk;
    float e = 0.f;
#pragma unroll
    for (int c = 0; c < 10; ++c) {
      float p = z[c] - tau;
      p = (p > 0.f) ? p : 0.f;
      distr[row * KSEL + c] = p;
      if (p > 0.f) e += p * logf(p);
    }
    atomicAdd(ent, -e * (1.f / (float)BATCH));
#pragma unroll
    for (int c = 0; c < 10; ++c) {
      meta[row * 32 + c]      = pos[c];
      meta[row * 32 + 10 + c] = (int)(best[c] & 0x3ffu);
    }
  }
}

// One block per row: stream logits once, write 10 candidate bit-vectors with
// the (<=10) flip positions patched inline. Nontemporal stores: 671MB >> L2.
__global__ void __launch_bounds__(256) bv_write_kernel(const float* __restrict__ logits,
                                                       const int* __restrict__ meta,
                                                       float* __restrict__ bv) {
  const int b = blockIdx.x;
  const float* __restrict__ r = logits + (size_t)b * LATENT;
  int pos[10], msk[10];
#pragma unroll
  for (int c = 0; c < 10; ++c) {
    pos[c] = meta[b * 32 + c];
    msk[c] = meta[b * 32 + 10 + c];
  }
  float* __restrict__ ob = bv + (size_t)b * (KSEL * LATENT);
#pragma unroll
  for (int pass = 0; pass < LATENT / (256 * 4); ++pass) {
    const int j = (pass * 256 + (int)threadIdx.x) * 4;
    const v4f x = *reinterpret_cast<const v4f*>(r + j);
    const float b0 = (x.x > 0.f) ? 1.f : 0.f;
    const float b1 = (x.y > 0.f) ? 1.f : 0.f;
    const float b2 = (x.z > 0.f) ? 1.f : 0.f;
    const float b3 = (x.w > 0.f) ? 1.f : 0.f;
    int t0 = -1, t1 = -1, t2 = -1, t3 = -1;
#pragma unroll
    for (int c = 0; c < 10; ++c) {
      t0 = (pos[c] == j + 0) ? c : t0;
      t1 = (pos[c] == j + 1) ? c : t1;
      t2 = (pos[c] == j + 2) ? c : t2;
      t3 = (pos[c] == j + 3) ? c : t3;
    }
    const bool any = (t0 >= 0) || (t1 >= 0) || (t2 >= 0) || (t3 >= 0);
#pragma unroll
    for (int c = 0; c < 10; ++c) {
      v4f o;
      o.x = b0; o.y = b1; o.z = b2; o.w = b3;
      if (any) {
        if (t0 >= 0 && ((msk[c] >> t0) & 1)) o.x = 1.f - b0;
        if (t1 >= 0 && ((msk[c] >> t1) & 1)) o.y = 1.f - b1;
        if (t2 >= 0 && ((msk[c] >> t2) & 1)) o.z = 1.f - b2;
        if (t3 >= 0 && ((msk[c] >> t3) & 1)) o.w = 1.f - b3;
      }
      __builtin_nontemporal_store(o, reinterpret_cast<v4f*>(ob + (size_t)c * LATENT + j));
    }
  }
}

extern "C" void kernel_launch(void* const* d_in, const int* in_sizes, int n_in,
                              void* d_out, int out_size, void* d_ws, size_t ws_size,
                              hipStream_t stream) {
  (void)in_sizes; (void)n_in; (void)out_size; (void)ws_size;
  const float* logits = (const float*)d_in[0];
  float* out = (float*)d_out;

  const size_t BV = (size_t)BATCH * KSEL * LATENT;  // bv elements
  float* bv    = out;
  float* distr = out + BV;
  float* ent   = out + BV + (size_t)BATCH * KSEL;
  int*   meta  = (int*)d_ws;  // 4096 rows * 32 ints = 512 KB

  hipMemsetAsync(ent, 0, sizeof(float), stream);
  topk_rows_kernel<<<BATCH / 8, 256, 0, stream>>>(logits, distr, ent, meta);
  bv_write_kernel<<<BATCH, 256, 0, stream>>>(logits, meta, bv);
}